// Warp_36189394436782
// MI455X (gfx1250) — compile-verified
//
#include <hip/hip_runtime.h>
#include <hip/hip_bf16.h>

typedef float v4f __attribute__((ext_vector_type(4)));

namespace {
constexpr int Wd = 256;
constexpr int Hd = 256;
constexpr int Cd = 64;
constexpr int HWd = Wd * Hd;
constexpr int C_PER = 16;            // channels per block (gridDim.y = 4)
constexpr int PIX_PER_BLOCK = 1024;  // 4 rows of 256 pixels
}

__global__ __launch_bounds__(256) void warp_bilinear_kernel(
    const float* __restrict__ frame, const float* __restrict__ flow,
    float* __restrict__ out)
{
    __shared__ __align__(16) float sflow[2 * PIX_PER_BLOCK]; // fx rows, then fy rows

    const int tid = threadIdx.x;
    const long long pixbase = (long long)blockIdx.x * PIX_PER_BLOCK;
    const int n   = (int)(pixbase / HWd);
    const int rem = (int)(pixbase - (long long)n * HWd);
    const int y0  = rem / Wd;   // first of 4 consecutive rows handled by this block

    // ---- stage flow rows into LDS via CDNA5 async global->LDS copies ----
    const float* fxg = flow + (size_t)n * 2 * HWd + (size_t)y0 * Wd;
    const float* fyg = fxg + HWd;
    const unsigned ldsbase = (unsigned)(size_t)(&sflow[0]);
    const unsigned ldsA = ldsbase + (unsigned)tid * 16u;
    const unsigned ldsB = ldsA + (unsigned)PIX_PER_BLOCK * 4u;
    const unsigned long long ga = (unsigned long long)(size_t)fxg + (unsigned long long)tid * 16u;
    const unsigned long long gb = (unsigned long long)(size_t)fyg + (unsigned long long)tid * 16u;
    asm volatile("global_load_async_to_lds_b128 %0, %1, off" :: "v"(ldsA), "v"(ga) : "memory");
    asm volatile("global_load_async_to_lds_b128 %0, %1, off" :: "v"(ldsB), "v"(gb) : "memory");
    asm volatile("s_wait_asynccnt 0" ::: "memory");

    // each thread reads back exactly the 4 fx / 4 fy values it staged
    const v4f fx4 = *(const v4f*)(&sflow[tid * 4]);
    const v4f fy4 = *(const v4f*)(&sflow[PIX_PER_BLOCK + tid * 4]);

    const int y  = y0 + (tid >> 6);
    const int x0 = (tid & 63) * 4;

    float wgt[4][4];
    int   off[4][4];
    const float fxv[4] = {fx4.x, fx4.y, fx4.z, fx4.w};
    const float fyv[4] = {fy4.x, fy4.y, fy4.z, fy4.w};

#pragma unroll
    for (int j = 0; j < 4; ++j) {
        const int x = x0 + j;
        // replicate reference arithmetic (fp32, same op order)
        const float gx = ((float)x + 0.5f) * (2.0f / Wd) - 1.0f;
        const float gy = ((float)y + 0.5f) * (2.0f / Hd) - 1.0f;
        const float xn = gx + fxv[j] * (1.0f / Wd);
        const float yn = gy + fyv[j] * (1.0f / Hd);
        const float ixf = ((xn + 1.0f) * (float)Wd - 1.0f) * 0.5f;
        const float iyf = ((yn + 1.0f) * (float)Hd - 1.0f) * 0.5f;
        const float ix0f = floorf(ixf);
        const float iy0f = floorf(iyf);
        const float wx1 = ixf - ix0f, wy1 = iyf - iy0f;
        const float wx0 = 1.0f - wx1, wy0 = 1.0f - wy1;
        const int ix0 = (int)ix0f, iy0 = (int)iy0f;
        const int ix1 = ix0 + 1,   iy1 = iy0 + 1;
        // zero-padding: fold validity masks into the weights; addresses are clamped
        const float mx0 = (ix0 >= 0 && ix0 < Wd) ? 1.0f : 0.0f;
        const float mx1 = (ix1 >= 0 && ix1 < Wd) ? 1.0f : 0.0f;
        const float my0 = (iy0 >= 0 && iy0 < Hd) ? 1.0f : 0.0f;
        const float my1 = (iy1 >= 0 && iy1 < Hd) ? 1.0f : 0.0f;
        const int cx0 = min(max(ix0, 0), Wd - 1);
        const int cx1 = min(max(ix1, 0), Wd - 1);
        const int cy0 = min(max(iy0, 0), Hd - 1);
        const int cy1 = min(max(iy1, 0), Hd - 1);
        wgt[j][0] = wy0 * wx0 * (my0 * mx0);
        wgt[j][1] = wy0 * wx1 * (my0 * mx1);
        wgt[j][2] = wy1 * wx0 * (my1 * mx0);
        wgt[j][3] = wy1 * wx1 * (my1 * mx1);
        off[j][0] = cy0 * Wd + cx0;
        off[j][1] = cy0 * Wd + cx1;
        off[j][2] = cy1 * Wd + cx0;
        off[j][3] = cy1 * Wd + cx1;
    }

    const int cbeg = (int)blockIdx.y * C_PER;
    const size_t planebase = ((size_t)n * Cd + cbeg) * (size_t)HWd;
    const float* __restrict__ ip = frame + planebase;
    float* __restrict__ op = out + planebase + (size_t)y * Wd + (size_t)x0;

#pragma unroll 4
    for (int c = 0; c < C_PER; ++c) {
        const float* __restrict__ p = ip + (size_t)c * HWd;
        v4f r;
        r.x = wgt[0][0]*p[off[0][0]] + wgt[0][1]*p[off[0][1]] + wgt[0][2]*p[off[0][2]] + wgt[0][3]*p[off[0][3]];
        r.y = wgt[1][0]*p[off[1][0]] + wgt[1][1]*p[off[1][1]] + wgt[1][2]*p[off[1][2]] + wgt[1][3]*p[off[1][3]];
        r.z = wgt[2][0]*p[off[2][0]] + wgt[2][1]*p[off[2][1]] + wgt[2][2]*p[off[2][2]] + wgt[2][3]*p[off[2][3]];
        r.w = wgt[3][0]*p[off[3][0]] + wgt[3][1]*p[off[3][1]] + wgt[3][2]*p[off[3][2]] + wgt[3][3]*p[off[3][3]];
        // streaming output: non-temporal 16B store, keep L2 for gathered frame taps
        __builtin_nontemporal_store(r, (v4f*)(op + (size_t)c * HWd));
    }
}

extern "C" void kernel_launch(void* const* d_in, const int* in_sizes, int n_in,
                              void* d_out, int out_size, void* d_ws, size_t ws_size,
                              hipStream_t stream) {
    const float* frame = (const float*)d_in[0];  // (8,64,256,256) f32
    const float* flow  = (const float*)d_in[1];  // (8,2,256,256)  f32
    float* out = (float*)d_out;                  // (8,64,256,256) f32

    const int total_pixels = 8 * Hd * Wd;                 // 524288
    dim3 grid(total_pixels / PIX_PER_BLOCK, Cd / C_PER);  // (512, 4)
    warp_bilinear_kernel<<<grid, 256, 0, stream>>>(frame, flow, out);
}